// ONNX_ORT_39333310496770
// MI455X (gfx1250) — compile-verified
//
#include <hip/hip_runtime.h>

#define T_DET 100
#define NPRED 25200
#define CPRED 117
#define NCLS  80
#define NM    32
#define PH    160
#define PW    160
#define PIX   (PH*PW)          // 25600
#define OW    (7 + PIX)        // 25607 floats per output row
#define NBATCH 8
#define POOLER 0.25f

typedef __attribute__((ext_vector_type(2))) float v2f;
typedef __attribute__((ext_vector_type(8))) float v8f;

// ---------------------------------------------------------------------------
// Kernel 1: per-detection header + coef/box staging + per-batch ranges.
// Only 100 rows of preds are ever touched (gather-aware).
// ---------------------------------------------------------------------------
__global__ __launch_bounds__(128)
void prep_kernel(const float* __restrict__ preds,
                 const int*   __restrict__ sel_batch,
                 const int*   __restrict__ sel_idx,
                 float*       __restrict__ out,
                 float*       __restrict__ ws_coef,   // [100][32]
                 float*       __restrict__ ws_box,    // [100][4] scaled boxes
                 int*         __restrict__ ws_range)  // [8] start, [8] cnt
{
    int tid = threadIdx.x;
    if (tid < T_DET) {
        int b = sel_batch[tid];
        int y = sel_idx[tid];
        const float* p = preds + ((long long)b * NPRED + y) * CPRED;
        float cx = p[0], cy = p[1], w = p[2], h = p[3];
        float bx0 = cx - 0.5f * w, by0 = cy - 0.5f * h;
        float bx1 = cx + 0.5f * w, by1 = cy + 0.5f * h;
        float conf = p[4];
        float best = p[5] * conf; int bestc = 0;
        #pragma unroll 4
        for (int c = 1; c < NCLS; ++c) {
            float s = p[5 + c] * conf;
            if (s > best) { best = s; bestc = c; }   // strict > matches argmax tie rule
        }
        float* orow = out + (long long)tid * OW;
        orow[0] = (float)b;
        orow[1] = bx0; orow[2] = by0; orow[3] = bx1; orow[4] = by1;
        orow[5] = (float)bestc;
        orow[6] = best;
        #pragma unroll
        for (int k = 0; k < NM; ++k) ws_coef[tid * NM + k] = p[5 + NCLS + k];
        ws_box[tid * 4 + 0] = bx0 * POOLER;
        ws_box[tid * 4 + 1] = by0 * POOLER;
        ws_box[tid * 4 + 2] = bx1 * POOLER;
        ws_box[tid * 4 + 3] = by1 * POOLER;
    }
    // sel_batch is sorted: start[b] = #entries < b, cnt[b] = #entries == b
    if (tid >= T_DET && tid < T_DET + NBATCH) {
        int bb = tid - T_DET;
        int st = 0, ct = 0;
        for (int i = 0; i < T_DET; ++i) {
            int v = sel_batch[i];
            st += (v < bb);
            ct += (v == bb);
        }
        ws_range[bb] = st;
        ws_range[NBATCH + bb] = ct;
    }
}

// ---------------------------------------------------------------------------
// Kernel 2: masks = crop( sigmoid( coef[16 det] x proto[b][32][16 pix] ) )
// via 8 chained V_WMMA_F32_16X16X4_F32 (K=32). Block = 4 wave32s = 64 pixels.
// Grid: (400 pixel chunks, 7 det tiles, 8 batches); empty tiles exit uniformly.
// ---------------------------------------------------------------------------
__global__ __launch_bounds__(128)
void mask_wmma_kernel(const float* __restrict__ proto,
                      const float* __restrict__ ws_coef,
                      const float* __restrict__ ws_box,
                      const int*   __restrict__ ws_range,
                      float*       __restrict__ out)
{
    __shared__ float sbox[16 * 4];

    const int b = blockIdx.z;
    const int d = blockIdx.y;
    const int start = ws_range[b];
    const int cnt   = ws_range[NBATCH + b];
    if (d * 16 >= cnt) return;                       // block-uniform exit
    const int det_start = start + d * 16;
    int rem = cnt - d * 16; if (rem > 16) rem = 16;  // valid rows in this tile

    const int tid  = threadIdx.x;
    if (tid < 64) {                                  // stage 16 scaled boxes in LDS
        int m = tid >> 2, comp = tid & 3;
        int t = det_start + m;
        sbox[tid] = (m < rem) ? ws_box[t * 4 + comp] : 0.0f;
    }
    __syncthreads();

    const int lane  = tid & 31;
    const int wave  = tid >> 5;
    const int mlane = lane & 15;
    const int hi    = lane >> 4;                     // lane half selects K pair
    const int pix   = blockIdx.x * 64 + wave * 16 + mlane;

    // --- A matrix: 16x32 coefs, rows past `rem` forced to zero (no divergence)
    const bool validA = (mlane < rem);
    const float* coefrow = ws_coef + (size_t)(validA ? (det_start + mlane) : det_start) * NM;
    float a0[8], a1[8];
    #pragma unroll
    for (int s = 0; s < 8; ++s) {
        int k = 4 * s + 2 * hi;
        float v0 = coefrow[k], v1 = coefrow[k + 1];
        a0[s] = validA ? v0 : 0.0f;
        a1[s] = validA ? v1 : 0.0f;
    }

    // --- B matrix: proto[b][k][pix], 64B-contiguous per half-wave per K row
    const float* pb = proto + (long long)b * NM * PIX;
    v8f c = {0.f, 0.f, 0.f, 0.f, 0.f, 0.f, 0.f, 0.f};
    #pragma unroll
    for (int s = 0; s < 8; ++s) {
        int k = 4 * s + 2 * hi;
        v2f A;  A.x  = a0[s];              A.y  = a1[s];
        v2f Bm; Bm.x = pb[(size_t)k * PIX + pix];
                Bm.y = pb[(size_t)(k + 1) * PIX + pix];
        // D = A(16x4,f32) * B(4x16,f32) + C(16x16,f32)
        c = __builtin_amdgcn_wmma_f32_16x16x4_f32(
                /*neg_a=*/false, A, /*neg_b=*/false, Bm,
                /*c_mod=*/(short)0, c, /*reuse_a=*/false, /*reuse_b=*/false);
    }

    // --- sigmoid + crop + scatter-store (divergence only after WMMA chain)
    const float fx = (float)(pix % PW);
    const float fy = (float)(pix / PW);
    #pragma unroll
    for (int r = 0; r < 8; ++r) {
        int m = r + 8 * hi;                // C/D row layout: VGPR r -> M=r / r+8
        if (m < rem) {
            int t = det_start + m;
            float x1 = sbox[m * 4 + 0], y1 = sbox[m * 4 + 1];
            float x2 = sbox[m * 4 + 2], y2 = sbox[m * 4 + 3];
            bool inb = (fx >= x1) && (fx < x2) && (fy >= y1) && (fy < y2);
            float sig = 1.0f / (1.0f + __expf(-c[r]));
            out[(long long)t * OW + 7 + pix] = inb ? sig : 0.0f;
        }
    }
}

// ---------------------------------------------------------------------------
extern "C" void kernel_launch(void* const* d_in, const int* in_sizes, int n_in,
                              void* d_out, int out_size, void* d_ws, size_t ws_size,
                              hipStream_t stream)
{
    const float* preds     = (const float*)d_in[0];
    const float* proto     = (const float*)d_in[1];
    const int*   sel_batch = (const int*)d_in[2];
    const int*   sel_idx   = (const int*)d_in[3];
    float* out = (float*)d_out;

    float* ws_coef  = (float*)d_ws;                     // 100*32 floats
    float* ws_box   = ws_coef + T_DET * NM;             // 100*4 floats
    int*   ws_range = (int*)(ws_box + T_DET * 4);       // 16 ints

    prep_kernel<<<1, 128, 0, stream>>>(preds, sel_batch, sel_idx,
                                       out, ws_coef, ws_box, ws_range);

    dim3 grid(PIX / 64, 7, NBATCH);                     // 400 x 7 x 8
    mask_wmma_kernel<<<grid, 128, 0, stream>>>(proto, ws_coef, ws_box,
                                               ws_range, out);
}